// CubicalModel_ISM_56770877718629
// MI455X (gfx1250) — compile-verified
//
#include <hip/hip_runtime.h>

// CDNA5 / gfx1250: wave32, V_WMMA_F32_16X16X4_F32 fp32 matrix pipe.
typedef __attribute__((ext_vector_type(2))) float v2f;  // A/B operand: 2 VGPRs/lane
typedef __attribute__((ext_vector_type(8))) float v8f;  // C/D operand: 8 VGPRs/lane

#define PDIM   50000   // length of p / row length of I,J
#define NOUT   200     // dgm1 (100) ++ dgm2 (100)
#define KC     400     // K-chunk per block; PDIM/KC = 125 chunks; KC % 8 == 0
#define NCHUNK (PDIM / KC)
#define NTILE  13      // ceil(208/16): 13 tiles of 16 output rows (padded)

__global__ __launch_bounds__(256) void zero_out_kernel(float* out, int n) {
  int i = blockIdx.x * blockDim.x + threadIdx.x;
  if (i < n) out[i] = 0.0f;
}

// One wave (32 lanes) per block. blockIdx.x = output tile (16 rows),
// blockIdx.y = K-chunk. Each step does D(16x16) += A(16x4) * B(4x16) with
// B[k][n] = p[k] broadcast, so D[:,0] accumulates the 16 dot products.
__global__ __launch_bounds__(32) void cubical_gather_matvec_wmma(
    const float* __restrict__ p,
    const float* __restrict__ I,
    const float* __restrict__ J,
    const int*   __restrict__ inds1,
    const int*   __restrict__ inds2,
    float*       __restrict__ out) {
  const int lane   = threadIdx.x;        // 0..31 (wave32)
  const int tile   = blockIdx.x;         // 0..NTILE-1
  const int kchunk = blockIdx.y;         // 0..NCHUNK-1
  const int m      = lane & 15;          // A-matrix row handled by this lane
  const int khalf  = (lane >> 4) << 1;   // lanes 0-15 hold K={0,1}; 16-31 hold K={2,3}
  const int g      = tile * 16 + m;      // global output index (0..207; >=200 is pad)

  // Gather-first: resolve the image row this output needs (flat = r*28 + c).
  const float* rowptr = I;               // safe dummy for padding lanes
  if (g < 100) {
    const int flat = inds1[2 * g] * 28 + inds1[2 * g + 1];
    rowptr = I + (size_t)flat * PDIM;
  } else if (g < NOUT) {
    const int gg   = g - 100;
    const int flat = inds2[2 * gg] * 28 + inds2[2 * gg + 1];
    rowptr = J + (size_t)flat * PDIM;
  }

  const int    k0   = kchunk * KC;
  const float* arow = rowptr + k0 + khalf;  // 8B-aligned: k0,khalf even
  const float* brow = p      + k0 + khalf;  // broadcast within half-wave

  // Two accumulators break the WMMA->WMMA RAW dependency chain on C/D.
  v8f acc0 = {};
  v8f acc1 = {};
#pragma unroll 4
  for (int k = 0; k < KC; k += 8) {
    v2f a0 = *(const v2f*)(arow + k);
    v2f b0 = *(const v2f*)(brow + k);
    v2f a1 = *(const v2f*)(arow + k + 4);
    v2f b1 = *(const v2f*)(brow + k + 4);
    // 8 args: (neg_a, A, neg_b, B, c_mod, C, reuse_a, reuse_b)
    acc0 = __builtin_amdgcn_wmma_f32_16x16x4_f32(
        false, a0, false, b0, (short)0, acc0, false, false);
    acc1 = __builtin_amdgcn_wmma_f32_16x16x4_f32(
        false, a1, false, b1, (short)0, acc1, false, false);
  }

  // D column 0 lives in lane 0 (rows 0..7) and lane 16 (rows 8..15).
  if ((lane & 15) == 0) {
    const int mbase = (lane >> 4) * 8;
#pragma unroll
    for (int v = 0; v < 8; ++v) {
      const int go = tile * 16 + mbase + v;
      if (go < NOUT) atomicAdd(&out[go], acc0[v] + acc1[v]);
    }
  }
}

extern "C" void kernel_launch(void* const* d_in, const int* in_sizes, int n_in,
                              void* d_out, int out_size, void* d_ws, size_t ws_size,
                              hipStream_t stream) {
  const float* p     = (const float*)d_in[0];   // [50000]
  const float* I     = (const float*)d_in[1];   // [784, 50000]
  const float* J     = (const float*)d_in[2];   // [784, 50000]
  const int*   inds1 = (const int*)d_in[3];     // [100, 2]
  const int*   inds2 = (const int*)d_in[4];     // [100, 2]
  float*       out   = (float*)d_out;           // [200] = dgm1 ++ dgm2 (flat)

  // Output is accumulated with atomics across K-chunks -> zero it first
  // (stream-ordered; graph-capture safe; deterministic work every call).
  zero_out_kernel<<<1, 256, 0, stream>>>(out, out_size);

  dim3 grid(NTILE, NCHUNK);   // 13 x 125 = 1625 single-wave blocks
  cubical_gather_matvec_wmma<<<grid, 32, 0, stream>>>(p, I, J, inds1, inds2, out);
}